// NonLinearEKV_Conv2d_28716151341200
// MI455X (gfx1250) — compile-verified
//
#include <hip/hip_runtime.h>
#include <cstdint>

#define AS1 __attribute__((address_space(1)))
#define AS3 __attribute__((address_space(3)))

typedef int v4i_t __attribute__((vector_size(16)));   // matches builtin's 'V4i' param

namespace {
constexpr int Bn    = 4;
constexpr int CIN   = 64;
constexpr int Hn    = 32;
constexpr int Wn    = 32;
constexpr int COUT  = 128;
constexpr int ROWS  = 4;                  // output rows per block
constexpr int CO_BLK = 16;                // output channels per block
constexpr int VROWS = ROWS + 2;           // 6 rows incl. halo
constexpr int VCOLS = 40;                 // cols 3..36 used (gx = -1..32); 40*4B = 160B row -> 16B aligned
constexpr int THN_ELEMS = CO_BLK * CIN * 9;      // 9216 floats (36 KB)
constexpr int VT_ELEMS  = CIN * VROWS * VCOLS;   // 15360 floats (60 KB)
constexpr int LDS_BYTES = (THN_ELEMS + VT_ELEMS) * 4;   // 98304 B (< 320KB/WGP)

constexpr double INVd  = 1.0 / 0.075;                    // 1/(2*N*Vt)
constexpr double LOG2E = 1.4426950408889634;
constexpr double LN2   = 0.6931471805599453;
constexpr float  Sf     = (float)(INVd * LOG2E);         // x = (V - th) * Sf  ->  2^x = exp((V-th)*inv)
constexpr float  K2f    = 0.26359713811572677f;          // exp(-VD*inv) = exp(-4/3)
constexpr float  FSCALE = (float)(0.0005625 * LN2 * LN2);// EKV_ALPHA * ln2^2
}

// ---- CDNA5 async global->LDS DMA (ASYNCcnt path) -------------------------
__device__ __forceinline__ void async_copy_b128(const float* g, float* l) {
#if __has_builtin(__builtin_amdgcn_global_load_async_to_lds_b128)
    __builtin_amdgcn_global_load_async_to_lds_b128((AS1 v4i_t*)g, (AS3 v4i_t*)l, 0, 0);
#else
    unsigned loff = (unsigned)(uintptr_t)(AS3 char*)l;   // raw LDS byte offset
    asm volatile("global_load_async_to_lds_b128 %0, %1, off"
                 :: "v"(loff), "v"(g) : "memory");
#endif
}

__device__ __forceinline__ void wait_async0() {
#if __has_builtin(__builtin_amdgcn_s_wait_asynccnt)
    __builtin_amdgcn_s_wait_asynccnt(0);
#else
    asm volatile("s_wait_asynccnt 0" ::: "memory");
#endif
}

__global__ __launch_bounds__(256)
void ekv_conv2d_kernel(const float* __restrict__ V_in,
                       const float* __restrict__ theta,
                       float* __restrict__ out) {
    extern __shared__ float smem[];
    float* thn = smem;               // [CO_BLK][CIN][9]  holds -theta * Sf
    float* vt  = smem + THN_ELEMS;   // [CIN][VROWS][VCOLS], zero-padded halo

    const int tid  = threadIdx.x;
    const int tile = blockIdx.x;     // 0..7  (row tile)
    const int cb   = blockIdx.y;     // 0..7  (co block)
    const int b    = blockIdx.z;     // 0..3
    const int py0  = tile * ROWS;

    // ---- 1) zero the V tile (provides the PAD=1 zero border) ------------
    for (int i = tid; i < VT_ELEMS / 4; i += 256)
        ((float4*)vt)[i] = make_float4(0.f, 0.f, 0.f, 0.f);
    __syncthreads();   // zeros visible before async DMA overwrites interior

    // ---- 2) async DMA: theta slice (contiguous 36KB) -> LDS --------------
    const float* tsrc = theta + (size_t)cb * CO_BLK * CIN * 9;
    for (int i = tid; i < THN_ELEMS / 4; i += 256)
        async_copy_b128(tsrc + i * 4, thn + i * 4);

    // ---- 3) async DMA: input tile interior rows -> LDS -------------------
    // per (cin,row): 32 floats = 8 x 16B segments; 64*6*8 = 3072 segments
    for (int i = tid; i < CIN * VROWS * 8; i += 256) {
        const int cin = i / (VROWS * 8);
        const int rem = i - cin * (VROWS * 8);
        const int r   = rem >> 3;           // 0..5
        const int seg = rem & 7;            // 0..7 (4 floats each)
        const int gy  = py0 - 1 + r;
        if (gy >= 0 && gy < Hn) {
            const float* g = V_in + (((size_t)(b * CIN + cin) * Hn + gy) * Wn) + seg * 4;
            float* l = vt + (cin * VROWS + r) * VCOLS + 4 + seg * 4;  // col 4 = gx 0 (16B aligned)
            async_copy_b128(g, l);
        }
    }
    wait_async0();
    __syncthreads();

    // ---- 4) pre-scale theta in place: thn = -theta * Sf ------------------
    for (int i = tid; i < THN_ELEMS; i += 256)
        thn[i] = thn[i] * (-Sf);
    __syncthreads();

    // ---- 5) compute ------------------------------------------------------
    const int pos = tid & 127;
    const int px  = pos & 31;        // lane-consecutive -> conflict-free LDS / coalesced stores
    const int py  = pos >> 5;        // 0..3
    const int cg  = tid >> 7;        // 0..1
    const int coL = cg * 8;          // local co base

    float acc[8];
#pragma unroll
    for (int c = 0; c < 8; ++c) acc[c] = 0.f;

    for (int cin = 0; cin < CIN; ++cin) {
        // 3x3 window around (py0+py, px); LDS col = gx + 4
        float v[9];
        const float* vr = vt + (cin * VROWS + py) * VCOLS + px + 3;
#pragma unroll
        for (int dy = 0; dy < 3; ++dy)
#pragma unroll
            for (int dx = 0; dx < 3; ++dx)
                v[dy * 3 + dx] = vr[dy * VCOLS + dx];

#pragma unroll
        for (int c = 0; c < 8; ++c) {
            const float* thc = thn + ((coL + c) * CIN + cin) * 9;  // wave-uniform -> LDS broadcast
            float a = acc[c];
#pragma unroll
            for (int t = 0; t < 9; ++t) {
                // x = (V - th)*inv*log2e ; e1 = exp((V-th)*inv) ; e2 = e1*exp(-VD*inv)
                float x  = __builtin_fmaf(v[t], Sf, thc[t]);
                float e1 = __builtin_amdgcn_exp2f(x);
                float e2 = e1 * K2f;
                float g1 = __builtin_amdgcn_logf(1.0f + e1);   // log2(1+e1) = softplus/ln2
                float g2 = __builtin_amdgcn_logf(1.0f + e2);
                a = __builtin_fmaf(g1 - g2, g1 + g2, a);       // (sp1^2 - sp2^2)/ln2^2
            }
            acc[c] = a;
        }
    }

    // ---- 6) store: out[b, co, gy, px] * alpha*ln2^2 ----------------------
    const int gy = py0 + py;
    float* op = out + (((size_t)(b * COUT + cb * CO_BLK + coL) * Hn + gy) * Wn) + px;
#pragma unroll
    for (int c = 0; c < 8; ++c)
        op[(size_t)c * Hn * Wn] = acc[c] * FSCALE;
}

extern "C" void kernel_launch(void* const* d_in, const int* in_sizes, int n_in,
                              void* d_out, int out_size, void* d_ws, size_t ws_size,
                              hipStream_t stream) {
    (void)in_sizes; (void)n_in; (void)d_ws; (void)ws_size; (void)out_size;
    const float* V_in  = (const float*)d_in[0];   // (4, 64, 32, 32) f32
    const float* theta = (const float*)d_in[1];   // (128, 64, 3, 3) f32
    float* out = (float*)d_out;                   // (4, 128, 32, 32) f32

    dim3 grid(Hn / ROWS, COUT / CO_BLK, Bn);      // (8, 8, 4)
    dim3 block(256);
    ekv_conv2d_kernel<<<grid, block, LDS_BYTES, stream>>>(V_in, theta, out);
}